// _BatchNorm_56968446214642
// MI455X (gfx1250) — compile-verified
//
#include <hip/hip_runtime.h>
#include <hip/hip_bf16.h>
#include <math.h>

// Shapes (fixed by the reference): x is (B=16, C=64, H=256, W=256) f32, NCHW.
// D = 16 quaternion channels; components r,i,j,k are channel blocks of 16.
#define HW     65536          // H*W
#define NBATCH 16
#define NCHAN  64
#define QD     16             // D
#define NPIX   (NBATCH * HW)  // elements per (d, component) = 1<<20
#define EPS    0.0001

typedef float vf4  __attribute__((ext_vector_type(4)));
typedef float v2f  __attribute__((ext_vector_type(2)));
typedef float v8f  __attribute__((ext_vector_type(8)));

// ---------------------------------------------------------------------------
// Kernel A: raw moments per quaternion channel d.
// Computes 14 sums over all B*H*W pixels: S_r,S_i,S_j,S_k and the 10 pairwise
// product sums (rr,ri,rj,rk,ii,ij,ik,jj,jk,kk). Two-stage, fixed-order
// reduction (deterministic). Grid: (64 blocks, D). Block: 256 threads.
// Each thread does 16 iterations of 4x float4 loads (one per component).
// ---------------------------------------------------------------------------
#define MOM_BLOCKS 64

__global__ __launch_bounds__(256)
void qbn_moments(const float* __restrict__ x, float* __restrict__ partials) {
    const int d   = blockIdx.y;
    const int bx  = blockIdx.x;
    const int tid = threadIdx.x;

    vf4 acc[14];
#pragma unroll
    for (int v = 0; v < 14; ++v) acc[v] = (vf4)0.0f;

    // float4-granular flat index over (B, HW/4) for this d: 1<<18 vec4s total,
    // 16384 threads per d -> exactly 16 iterations, coalesced.
#pragma unroll 4
    for (int it = 0; it < 16; ++it) {
        const int g   = it * 16384 + bx * 256 + tid;   // [0, 1<<18)
        const int b   = g >> 14;                        // batch
        const int hw4 = g & 16383;                      // vec4 index in plane
        const float* base = x + (((size_t)(b * NCHAN + d)) << 16) + (size_t)hw4 * 4;
        vf4 r = *(const vf4*)(base);
        vf4 i = *(const vf4*)(base + ((size_t)16 << 16));
        vf4 j = *(const vf4*)(base + ((size_t)32 << 16));
        vf4 k = *(const vf4*)(base + ((size_t)48 << 16));
        acc[0] += r;     acc[1] += i;     acc[2]  += j;     acc[3]  += k;
        acc[4] += r * r; acc[5] += r * i; acc[6]  += r * j; acc[7]  += r * k;
        acc[8] += i * i; acc[9] += i * j; acc[10] += i * k;
        acc[11]+= j * j; acc[12]+= j * k; acc[13] += k * k;
    }

    __shared__ float red[14 * 256];
#pragma unroll
    for (int v = 0; v < 14; ++v)
        red[v * 256 + tid] = acc[v][0] + acc[v][1] + acc[v][2] + acc[v][3];
    __syncthreads();
    for (int stride = 128; stride > 0; stride >>= 1) {
        if (tid < stride) {
#pragma unroll
            for (int v = 0; v < 14; ++v)
                red[v * 256 + tid] += red[v * 256 + tid + stride];
        }
        __syncthreads();
    }
    if (tid < 14)
        partials[(size_t)(d * MOM_BLOCKS + bx) * 16 + tid] = red[tid * 256];
}

// ---------------------------------------------------------------------------
// Kernel B: per-d finalize (tiny). Fixed-order f64 accumulation of the 64
// block partials, mean/cov, 4x4 Cholesky entries exactly as the reference,
// then compose T = Sym(gamma) * Sym(cholW) and bias = beta - T*mu.
// Output: coef[d*20 + 0..15] = T row-major, coef[d*20 + 16..19] = bias.
// ---------------------------------------------------------------------------
__global__ void qbn_finalize(const float* __restrict__ partials,
                             const float* __restrict__ g_rr, const float* __restrict__ g_ri,
                             const float* __restrict__ g_rj, const float* __restrict__ g_rk,
                             const float* __restrict__ g_ii, const float* __restrict__ g_ij,
                             const float* __restrict__ g_ik, const float* __restrict__ g_jj,
                             const float* __restrict__ g_jk, const float* __restrict__ g_kk,
                             const float* __restrict__ beta,
                             float* __restrict__ coef) {
    const int d = threadIdx.x;
    if (d >= QD) return;

    double a[14];
#pragma unroll
    for (int v = 0; v < 14; ++v) a[v] = 0.0;
    for (int blk = 0; blk < MOM_BLOCKS; ++blk) {
        const float* p = partials + (size_t)(d * MOM_BLOCKS + blk) * 16;
#pragma unroll
        for (int v = 0; v < 14; ++v) a[v] += (double)p[v];
    }

    const double N = (double)NPIX;
    const double mr = a[0] / N, mi = a[1] / N, mj = a[2] / N, mk = a[3] / N;
    const double Vrr = a[4]  / N - mr * mr + EPS;
    const double Vri = a[5]  / N - mr * mi;
    const double Vrj = a[6]  / N - mr * mj;
    const double Vrk = a[7]  / N - mr * mk;
    const double Vii = a[8]  / N - mi * mi + EPS;
    const double Vij = a[9]  / N - mi * mj;
    const double Vik = a[10] / N - mi * mk;
    const double Vjj = a[11] / N - mj * mj + EPS;
    const double Vjk = a[12] / N - mj * mk;
    const double Vkk = a[13] / N - mk * mk + EPS;

    // Cholesky entries, same recurrence as the reference.
    const double Wrr = sqrt(Vrr);
    const double Wri = Vri / Wrr;
    const double Wii = sqrt(Vii - Wri * Wri);
    const double Wrj = Vrj / Wrr;
    const double Wij = (Vij - Wri * Wrj) / Wii;
    const double Wjj = sqrt(Vjj - (Wij * Wij + Wrj * Wrj));
    const double Wrk = Vrk / Wrr;
    const double Wik = (Vik - Wri * Wrk) / Wii;
    const double Wjk = (Vjk - (Wij * Wik + Wrj * Wrk)) / Wjj;
    const double Wkk = sqrt(Vkk - (Wjk * Wjk + Wik * Wik + Wrk * Wrk));

    const double Wm[4][4] = {
        { Wrr, Wri, Wrj, Wrk },
        { Wri, Wii, Wij, Wik },
        { Wrj, Wij, Wjj, Wjk },
        { Wrk, Wik, Wjk, Wkk } };
    const double G[4][4] = {
        { (double)g_rr[d], (double)g_ri[d], (double)g_rj[d], (double)g_rk[d] },
        { (double)g_ri[d], (double)g_ii[d], (double)g_ij[d], (double)g_ik[d] },
        { (double)g_rj[d], (double)g_ij[d], (double)g_jj[d], (double)g_jk[d] },
        { (double)g_rk[d], (double)g_ik[d], (double)g_jk[d], (double)g_kk[d] } };

    const double mu[4] = { mr, mi, mj, mk };
    float* T = coef + d * 20;
#pragma unroll
    for (int m = 0; m < 4; ++m) {
        double bsum = (double)beta[m * QD + d];
#pragma unroll
        for (int n = 0; n < 4; ++n) {
            double t = 0.0;
#pragma unroll
            for (int kk2 = 0; kk2 < 4; ++kk2) t += G[m][kk2] * Wm[kk2][n];
            T[m * 4 + n] = (float)t;
            bsum -= t * mu[n];
        }
        T[16 + m] = (float)bsum;
    }
}

// ---------------------------------------------------------------------------
// Kernel C: streaming apply via V_WMMA_F32_16X16X4_F32.
//   D(16x16) = A(16x4) * B(4x16) + C
//   A = T (rows 0..3, rest zero).  ISA A(16x4) layout: lane l<16 -> row l,
//     v0=K0, v1=K1; lane l>=16 -> row l-16, v0=K2, v1=K3.
//   B = 16 pixels; B[k][n] = x_comp_k[pixel n].  Layout (documented B pattern,
//     K split across lane halves): v0 = K0 (lanes 0-15) / K2 (lanes 16-31),
//     v1 = K1 / K3  ->  v0 loads r|j planes, v1 loads i|k planes.
//   C = bias broadcast: VGPR m, lanes 0-15 (rows 0..7) -> bias[m] for m<4.
//   D row m, lanes 0-15 = output component m for the 16 pixels.
// Each wave does 2 WMMAs (32 pixels) per iteration; half-wave 64B stores.
// ---------------------------------------------------------------------------
__global__ __launch_bounds__(256)
void qbn_apply(const float* __restrict__ x, const float* __restrict__ coef,
               float* __restrict__ out) {
    const int d    = blockIdx.y;
    const int b    = blockIdx.z;
    const int tid  = threadIdx.x;
    const int lane = tid & 31;
    const int wave = tid >> 5;
    const int row  = lane & 15;
    const bool lo  = lane < 16;

    const float* T    = coef + d * 20;
    const float* bias = T + 16;

    v2f afrag = (v2f)0.0f;
    if (row < 4) {
        afrag[0] = lo ? T[row * 4 + 0] : T[row * 4 + 2];
        afrag[1] = lo ? T[row * 4 + 1] : T[row * 4 + 3];
    }
    v8f cfrag = (v8f)0.0f;
    if (lo) { cfrag[0] = bias[0]; cfrag[1] = bias[1]; cfrag[2] = bias[2]; cfrag[3] = bias[3]; }

    const size_t planeBase = ((size_t)(b * NCHAN + d)) << 16;   // r-plane of (b,d)
    // Per-lane component-plane pointers for the B fragments (lane index folded in).
    const float* pB0 = x + planeBase + (lo ? (size_t)0        : ((size_t)32 << 16)) + row; // r | j
    const float* pB1 = x + planeBase + (lo ? ((size_t)16 << 16) : ((size_t)48 << 16)) + row; // i | k
    float* oR = out + planeBase;
    float* oI = oR + ((size_t)16 << 16);
    float* oJ = oR + ((size_t)32 << 16);
    float* oK = oR + ((size_t)48 << 16);

    const int chunk = blockIdx.x * 4096 + wave * 512;           // 16 iters * 32 px
#pragma unroll 2
    for (int it = 0; it < 16; ++it) {
        const int p0 = chunk + it * 32;
        v2f bf1, bf2;
        bf1[0] = pB0[p0];       bf1[1] = pB1[p0];
        bf2[0] = pB0[p0 + 16];  bf2[1] = pB1[p0 + 16];
        v8f d1 = __builtin_amdgcn_wmma_f32_16x16x4_f32(false, afrag, false, bf1,
                                                       (short)0, cfrag, false, false);
        v8f d2 = __builtin_amdgcn_wmma_f32_16x16x4_f32(false, afrag, false, bf2,
                                                       (short)0, cfrag, false, false);
        if (lo) {
            const int p = p0 + row;
            oR[p]      = d1[0]; oI[p]      = d1[1]; oJ[p]      = d1[2]; oK[p]      = d1[3];
            oR[p + 16] = d2[0]; oI[p + 16] = d2[1]; oJ[p + 16] = d2[2]; oK[p + 16] = d2[3];
        }
    }
}

// ---------------------------------------------------------------------------
extern "C" void kernel_launch(void* const* d_in, const int* in_sizes, int n_in,
                              void* d_out, int out_size, void* d_ws, size_t ws_size,
                              hipStream_t stream) {
    const float* x    = (const float*)d_in[0];
    const float* grr  = (const float*)d_in[1];
    const float* gri  = (const float*)d_in[2];
    const float* grj  = (const float*)d_in[3];
    const float* grk  = (const float*)d_in[4];
    const float* gii  = (const float*)d_in[5];
    const float* gij  = (const float*)d_in[6];
    const float* gik  = (const float*)d_in[7];
    const float* gjj  = (const float*)d_in[8];
    const float* gjk  = (const float*)d_in[9];
    const float* gkk  = (const float*)d_in[10];
    const float* beta = (const float*)d_in[11];
    float* out = (float*)d_out;

    float* partials = (float*)d_ws;                        // 16*64*16 floats = 64 KiB
    float* coefs    = partials + QD * MOM_BLOCKS * 16;     // 16*20 floats

    qbn_moments<<<dim3(MOM_BLOCKS, QD), 256, 0, stream>>>(x, partials);
    qbn_finalize<<<1, 32, 0, stream>>>(partials, grr, gri, grj, grk, gii, gij,
                                       gik, gjj, gjk, gkk, beta, coefs);
    qbn_apply<<<dim3(HW / 4096, QD, NBATCH), 256, 0, stream>>>(x, coefs, out);
}